// NodeFeatures_1580547965174
// MI455X (gfx1250) — compile-verified
//
#include <hip/hip_runtime.h>

// NodeFeatures: out[b,i,h] = (x@U_w^T + U_b)[b,i,h]
//                          + (sum_j eg[b,i,j,h]*Vx[b,j,h]) / (1e-20 + sum_j eg[b,i,j,h])
// B=8, N=256, H=256.  edge_gate is 512MB f32 streamed once -> bandwidth bound.
// Kernel 1 does both Linears with V_WMMA_F32_16X16X4_F32 (f32-exact, tiny cost).
// Kernel 2 streams edge_gate with non-temporal loads and fuses everything.

typedef __attribute__((ext_vector_type(2))) float v2f;
typedef __attribute__((ext_vector_type(4))) float v4f;
typedef __attribute__((ext_vector_type(8))) float v8f;

#define Hdim 256
#define NROW 2048   // B*N flattened rows

// ---------------------------------------------------------------------------
// Kernel 1: C[m,k] = sum_h x[m,h]*W[k,h] + b[k]   (torch Linear, x @ W^T + b)
// One wave per 16x16 output tile. 4096 tiles total (2048 per matrix; U then V).
// ---------------------------------------------------------------------------
__global__ __launch_bounds__(256) void nf_linear_wmma(
    const float* __restrict__ x,
    const float* __restrict__ Uw, const float* __restrict__ Ub,
    const float* __restrict__ Vw, const float* __restrict__ Vb,
    float* __restrict__ outU, float* __restrict__ outV)
{
    const int widx = blockIdx.x * (blockDim.x >> 5) + (threadIdx.x >> 5);
    const int lane = threadIdx.x & 31;

    const int mat  = widx >> 11;      // 0 -> U, 1 -> V   (2048 tiles each)
    const int tile = widx & 2047;
    const int tm   = (tile >> 4) << 4;   // row tile base   (0..2032)
    const int tn   = (tile & 15) << 4;   // col tile base   (0..240)

    const float* __restrict__ W    = mat ? Vw : Uw;
    const float* __restrict__ bias = mat ? Vb : Ub;
    float* __restrict__ outp       = mat ? outV : outU;

    const int lhalf = lane >> 4;      // 0: K+{0,1}, 1: K+{2,3}
    const int l15   = lane & 15;

    // A: x rows tm..tm+15, K-contiguous.  B: W rows tn..tn+15 (output channels),
    // K-contiguous — identical lane mapping since both are row-major over K.
    const float* arow = x + (size_t)(tm + l15) * Hdim + 2 * lhalf;
    const float* brow = W + (size_t)(tn + l15) * Hdim + 2 * lhalf;

    // Seed accumulator with bias: C[M,N] layout has N = l15 in every VGPR.
    const float bv = bias[tn + l15];
    v8f c;
#pragma unroll
    for (int v = 0; v < 8; ++v) c[v] = bv;

#pragma unroll 8
    for (int k = 0; k < Hdim; k += 4) {
        v2f a = *(const v2f*)(arow + k);
        v2f b = *(const v2f*)(brow + k);
        // (neg_a, A, neg_b, B, c_mod, C, reuse_a, reuse_b)
        c = __builtin_amdgcn_wmma_f32_16x16x4_f32(
                false, a, false, b, (short)0, c, false, false);
    }

    // D layout: VGPR v -> row tm + v + 8*lhalf, col tn + l15.
    float* op = outp + (size_t)(tm + 8 * lhalf) * Hdim + tn + l15;
#pragma unroll
    for (int v = 0; v < 8; ++v) op[(size_t)v * Hdim] = c[v];
}

// ---------------------------------------------------------------------------
// Kernel 2: per (b,i) row: num4/den4 over j, then out = Ux + num/(den+1e-20).
// Block of 256 threads handles 4 rows; thread t owns float4 slice h4 = t&63
// of row r = 4*blockIdx.x + (t>>6). edge_gate read-once -> non-temporal.
// ---------------------------------------------------------------------------
__global__ __launch_bounds__(256) void nf_gate_agg(
    const float* __restrict__ eg,
    const float* __restrict__ Vx,
    float* __restrict__ out)
{
    const int t  = threadIdx.x;
    const int r  = (blockIdx.x << 2) + (t >> 6);   // flattened (b,i), 0..2047
    const int h4 = t & 63;                         // float4 index within H
    const int b  = r >> 8;

    const v4f* __restrict__ egp = (const v4f*)eg + ((size_t)r * 256) * 64 + h4;
    const v4f* __restrict__ vxp = (const v4f*)Vx + ((size_t)b * 256) * 64 + h4;

    v4f num = (v4f)0.0f;
    v4f den = (v4f)0.0f;
#pragma unroll 4
    for (int j = 0; j < 256; ++j) {
        v4f g = __builtin_nontemporal_load(egp + (size_t)j * 64);  // 512MB stream
        v4f v = vxp[(size_t)j * 64];                               // L2-resident
        num += g * v;
        den += g;
    }

    v4f* o = (v4f*)out + (size_t)r * 64 + h4;
    v4f u = *o;                                    // Ux written by kernel 1
    v4f res = u + num / (den + 1e-20f);
    __builtin_nontemporal_store(res, o);
}

// ---------------------------------------------------------------------------
extern "C" void kernel_launch(void* const* d_in, const int* in_sizes, int n_in,
                              void* d_out, int out_size, void* d_ws, size_t ws_size,
                              hipStream_t stream)
{
    const float* x  = (const float*)d_in[0];   // (8,256,256)
    const float* eg = (const float*)d_in[1];   // (8,256,256,256)
    const float* Uw = (const float*)d_in[2];   // (256,256)
    const float* Ub = (const float*)d_in[3];   // (256,)
    const float* Vw = (const float*)d_in[4];   // (256,256)
    const float* Vb = (const float*)d_in[5];   // (256,)

    float* out = (float*)d_out;                // (8,256,256) = 2MB
    float* Vx  = (float*)d_ws;                 // needs 2048*256*4 = 2MB scratch

    // 4096 tiles (U + V), 8 waves/block -> 512 blocks.
    nf_linear_wmma<<<512, 256, 0, stream>>>(x, Uw, Ub, Vw, Vb, out, Vx);
    // 2048 rows, 4 rows/block -> 512 blocks.
    nf_gate_agg<<<512, 256, 0, stream>>>(eg, Vx, out);
}